// GINEBackbone_32401233281338
// MI455X (gfx1250) — compile-verified
//
#include <hip/hip_runtime.h>

#define N_NODES 50000
#define N_EDGES 500000

typedef float v2f __attribute__((ext_vector_type(2)));
typedef float v8f __attribute__((ext_vector_type(8)));

// ---------------------------------------------------------------- utilities
__global__ void zero_kernel(float* __restrict__ p, int n) {
  int i = blockIdx.x * blockDim.x + threadIdx.x;
  if (i < n) p[i] = 0.f;
}

// Copy W1 [ind,256] into zero-padded [128,256] scratch.
__global__ void padw1_kernel(const float* __restrict__ W1, float* __restrict__ W1p, int ind) {
  int i = blockIdx.x * blockDim.x + threadIdx.x;
  if (i >= 128 * 256) return;
  int r = i >> 8;
  W1p[i] = (r < ind) ? W1[i] : 0.f;
}

// ------------------------------------------------------------ edge phase
// One wave per edge: e = ea@eW+eb (eW/eb cached in LDS), m = relu(x[src]+e),
// atomic scatter-add into hbuf[dst] (128-wide, zero-padded).
__global__ __launch_bounds__(256) void edge_agg_kernel(
    const float* __restrict__ x, int xstride, int ind,
    const int* __restrict__ ei, const float* __restrict__ ea,
    const float* __restrict__ eW, const float* __restrict__ eb,
    float* __restrict__ hbuf) {
  __shared__ float sW[12 * 128];
  __shared__ float sB[128];
  for (int i = threadIdx.x; i < 12 * ind; i += blockDim.x) sW[i] = eW[i];
  for (int i = threadIdx.x; i < ind; i += blockDim.x) sB[i] = eb[i];
  __syncthreads();
  int lane = threadIdx.x & 31;
  int wave = blockIdx.x * (blockDim.x >> 5) + (threadIdx.x >> 5);
  int nw = gridDim.x * (blockDim.x >> 5);
  for (int e = wave; e < N_EDGES; e += nw) {
    int s = ei[e];
    int d = ei[N_EDGES + e];
    const float* eap = ea + (size_t)e * 12;
    if (e + nw < N_EDGES) __builtin_prefetch(ea + (size_t)(e + nw) * 12, 0, 0);
    float eav[12];
#pragma unroll
    for (int k = 0; k < 12; ++k) eav[k] = eap[k];
    const float* xs = x + (size_t)s * xstride;
    float* hd = hbuf + (size_t)d * 128;
    for (int j = lane; j < ind; j += 32) {
      float acc = sB[j];
#pragma unroll
      for (int k = 0; k < 12; ++k) acc += eav[k] * sW[k * ind + j];
      float m = xs[j] + acc;
      m = m > 0.f ? m : 0.f;
      atomicAdd(hd + j, m);
    }
  }
}

// hbuf += (1+eps) * x   (first ind columns; padding columns stay zero)
__global__ void combine_kernel(float* __restrict__ hbuf, const float* __restrict__ x,
                               int xstride, int ind, const float* __restrict__ epsP, int n) {
  int i = blockIdx.x * blockDim.x + threadIdx.x;
  if (i >= n) return;
  int r = i / ind, k = i - r * ind;
  float c = 1.f + epsP[0];
  hbuf[(size_t)r * 128 + k] += c * x[(size_t)r * xstride + k];
}

// ------------------------------------------------------------ WMMA GEMM 1
// T[50000,256] = hbuf[50000,128(pad)] @ W1pad[128,256] + b1
__global__ __launch_bounds__(256) void gemm1_wmma_kernel(
    const float* __restrict__ A, const float* __restrict__ Bm,
    const float* __restrict__ bias, float* __restrict__ Out, int Kpad) {
  const int lda = 128, ldb = 256, ldc = 256, Ntiles = 16;
  int lane = threadIdx.x & 31;
  int wave = threadIdx.x >> 5;
  int tile = blockIdx.x * 8 + wave;           // 3125*16 tiles / 8 waves = 6250 blocks, exact
  int mtile = tile / Ntiles, ntile = tile % Ntiles;
  int ml = lane & 15, half = lane >> 4;
  int row = mtile * 16 + ml;
  int col = ntile * 16 + ml;
  v8f acc = {};
  for (int kb = 0; kb < Kpad; kb += 4) {
    int ka = kb + 2 * half;
    v2f a, b;
    a.x = A[(size_t)row * lda + ka];
    a.y = A[(size_t)row * lda + ka + 1];
    b.x = Bm[(size_t)ka * ldb + col];
    b.y = Bm[(size_t)(ka + 1) * ldb + col];
    acc = __builtin_amdgcn_wmma_f32_16x16x4_f32(false, a, false, b, (short)0, acc,
                                                false, false);
  }
  float bv = bias[col];
#pragma unroll
  for (int r = 0; r < 8; ++r)
    Out[(size_t)(mtile * 16 + r + 8 * half) * ldc + col] = acc[r] + bv;
}

// ------------------------------------------------------------ WMMA GEMM 2
// U[50000,128] = relu(T*scale1+shift1) @ W2[256,128] + b2   (BN1 fused into A-loads)
__global__ __launch_bounds__(256) void gemm2_wmma_kernel(
    const float* __restrict__ T, const float* __restrict__ sc, const float* __restrict__ sh,
    const float* __restrict__ W2, const float* __restrict__ b2, float* __restrict__ U) {
  __shared__ float ssc[256], ssh[256];
  ssc[threadIdx.x] = sc[threadIdx.x];
  ssh[threadIdx.x] = sh[threadIdx.x];
  __syncthreads();
  const int lda = 256, ldb = 128, ldc = 128, Ntiles = 8;
  int lane = threadIdx.x & 31;
  int wave = threadIdx.x >> 5;
  int tile = blockIdx.x * 8 + wave;           // 3125*8 tiles / 8 = 3125 blocks, exact
  int mtile = tile / Ntiles, ntile = tile % Ntiles;
  int ml = lane & 15, half = lane >> 4;
  int row = mtile * 16 + ml;
  int col = ntile * 16 + ml;
  v8f acc = {};
  for (int kb = 0; kb < 256; kb += 4) {
    int ka = kb + 2 * half;
    float t0 = T[(size_t)row * lda + ka] * ssc[ka] + ssh[ka];
    float t1 = T[(size_t)row * lda + ka + 1] * ssc[ka + 1] + ssh[ka + 1];
    v2f a, b;
    a.x = t0 > 0.f ? t0 : 0.f;
    a.y = t1 > 0.f ? t1 : 0.f;
    b.x = W2[(size_t)ka * ldb + col];
    b.y = W2[(size_t)(ka + 1) * ldb + col];
    acc = __builtin_amdgcn_wmma_f32_16x16x4_f32(false, a, false, b, (short)0, acc,
                                                false, false);
  }
  float bv = b2[col];
#pragma unroll
  for (int r = 0; r < 8; ++r)
    U[(size_t)(mtile * 16 + r + 8 * half) * ldc + col] = acc[r] + bv;
}

// --------------------------------------------------- BatchNorm batch stats
// Column sums / sumsq with coalesced row reads; requires blockDim % C == 0.
__global__ void colstats_kernel(const float* __restrict__ T, int C,
                                float* __restrict__ sum, float* __restrict__ sq) {
  int tid = threadIdx.x;
  int c = tid % C;
  int roff = tid / C;
  int rstep = blockDim.x / C;
  int chunk = (N_NODES + gridDim.x - 1) / gridDim.x;
  int r0 = blockIdx.x * chunk;
  int r1 = r0 + chunk;
  if (r1 > N_NODES) r1 = N_NODES;
  float s = 0.f, q = 0.f;
  for (int r = r0 + roff; r < r1; r += rstep) {
    float v = T[(size_t)r * C + c];
    s += v;
    q += v * v;
  }
  atomicAdd(&sum[c], s);
  atomicAdd(&sq[c], q);
}

// scale = g * rsqrt(var+1e-5); shift = beta - mean*scale  (biased var, matches jnp)
__global__ void finalize_kernel(const float* __restrict__ sum, const float* __restrict__ sq,
                                const float* __restrict__ g, const float* __restrict__ be,
                                float* __restrict__ scale, float* __restrict__ shift,
                                int C, float invN) {
  int c = blockIdx.x * blockDim.x + threadIdx.x;
  if (c >= C) return;
  float mu = sum[c] * invN;
  float var = sq[c] * invN - mu * mu;
  float s = g[c] * rsqrtf(var + 1e-5f);
  scale[c] = s;
  shift[c] = be[c] - mu * s;
}

// out = relu(u*scale2+shift2)  (+ residual base + 0.3*out for the middle layer)
__global__ void apply_kernel(const float* __restrict__ u, const float* __restrict__ sc,
                             const float* __restrict__ sh, const float* __restrict__ res,
                             float alpha, int useres, float* __restrict__ outp, int n) {
  int i = blockIdx.x * blockDim.x + threadIdx.x;
  if (i >= n) return;
  int c = i & 127;
  float v = u[i] * sc[c] + sh[c];
  v = v > 0.f ? v : 0.f;
  outp[i] = useres ? (res[i] + alpha * v) : v;
}

// --------------------------------------------------------------- launcher
extern "C" void kernel_launch(void* const* d_in, const int* in_sizes, int n_in,
                              void* d_out, int out_size, void* d_ws, size_t ws_size,
                              hipStream_t stream) {
  (void)in_sizes; (void)n_in; (void)out_size; (void)ws_size;
  const float* x0 = (const float*)d_in[0];
  const int* ei = (const int*)d_in[1];
  const float* ea = (const float*)d_in[2];

  float* ws = (float*)d_ws;
  size_t off = 0;
  float* hbuf = ws + off; off += (size_t)N_NODES * 128;
  float* tbuf = ws + off; off += (size_t)N_NODES * 256;
  float* ubuf = ws + off; off += (size_t)N_NODES * 128;
  float* xA   = ws + off; off += (size_t)N_NODES * 128;
  float* xB   = ws + off; off += (size_t)N_NODES * 128;
  float* w1p  = ws + off; off += 128 * 256;
  float* sum1 = ws + off; off += 256;
  float* sq1  = ws + off; off += 256;
  float* sc1  = ws + off; off += 256;
  float* sh1  = ws + off; off += 256;
  float* sum2 = ws + off; off += 128;
  float* sq2  = ws + off; off += 128;
  float* sc2  = ws + off; off += 128;
  float* sh2  = ws + off; off += 128;

  for (int layer = 0; layer < 3; ++layer) {
    const int ind = (layer == 0) ? 27 : 128;
    const int Kpad = (layer == 0) ? 28 : 128;
    const int xstride = (layer == 0) ? 27 : 128;
    const float* xin = (layer == 0) ? x0 : (layer == 1 ? xA : xB);
    float* xnext = (layer == 0) ? xA : (layer == 1 ? xB : (float*)d_out);
    const int base = 3 + layer * 11;
    const float* epsP = (const float*)d_in[base + 0];
    const float* eW   = (const float*)d_in[base + 1];
    const float* eb   = (const float*)d_in[base + 2];
    const float* W1   = (const float*)d_in[base + 3];
    const float* b1   = (const float*)d_in[base + 4];
    const float* g1   = (const float*)d_in[base + 5];
    const float* be1  = (const float*)d_in[base + 6];
    const float* W2   = (const float*)d_in[base + 7];
    const float* b2   = (const float*)d_in[base + 8];
    const float* ng   = (const float*)d_in[base + 9];
    const float* nb   = (const float*)d_in[base + 10];

    const int nh = N_NODES * 128;
    zero_kernel<<<(nh + 255) / 256, 256, 0, stream>>>(hbuf, nh);
    padw1_kernel<<<128, 256, 0, stream>>>(W1, w1p, ind);
    edge_agg_kernel<<<2048, 256, 0, stream>>>(xin, xstride, ind, ei, ea, eW, eb, hbuf);
    const int ncm = N_NODES * ind;
    combine_kernel<<<(ncm + 255) / 256, 256, 0, stream>>>(hbuf, xin, xstride, ind, epsP, ncm);
    gemm1_wmma_kernel<<<6250, 256, 0, stream>>>(hbuf, w1p, b1, tbuf, Kpad);
    zero_kernel<<<2, 256, 0, stream>>>(sum1, 512);   // sum1 & sq1 are contiguous
    colstats_kernel<<<512, 256, 0, stream>>>(tbuf, 256, sum1, sq1);
    finalize_kernel<<<1, 256, 0, stream>>>(sum1, sq1, g1, be1, sc1, sh1, 256, 1.f / N_NODES);
    gemm2_wmma_kernel<<<3125, 256, 0, stream>>>(tbuf, sc1, sh1, W2, b2, ubuf);
    zero_kernel<<<1, 256, 0, stream>>>(sum2, 256);   // sum2 & sq2 are contiguous
    colstats_kernel<<<512, 256, 0, stream>>>(ubuf, 128, sum2, sq2);
    finalize_kernel<<<1, 128, 0, stream>>>(sum2, sq2, ng, nb, sc2, sh2, 128, 1.f / N_NODES);
    apply_kernel<<<(nh + 255) / 256, 256, 0, stream>>>(ubuf, sc2, sh2, xin, 0.3f,
                                                       (layer == 1) ? 1 : 0, xnext, nh);
  }
}